// kGraphRNA_46093589020843
// MI455X (gfx1250) — compile-verified
//
#include <hip/hip_runtime.h>
#include <hip/hip_bf16.h>

typedef __attribute__((ext_vector_type(2))) float v2f;
typedef __attribute__((ext_vector_type(8))) float v8f;

#define FDIM 128   // feature dim F == hidden dim H
#define LDSP 132   // padded LDS row stride for A tiles (floats)
#define WSTR 130   // padded LDS row stride for transposed weights (floats)

// ---------------------------------------------------------------------------
// zero-fill scratch (agg + counts) -- grid-stride
// ---------------------------------------------------------------------------
__global__ void kZero(float* __restrict__ p, size_t n) {
    size_t i = (size_t)blockIdx.x * blockDim.x + threadIdx.x;
    size_t stride = (size_t)gridDim.x * blockDim.x;
    for (; i < n; i += stride) p[i] = 0.0f;
}

// ---------------------------------------------------------------------------
// segment-sum aggregation: one wave per edge, float4 per lane (32*4 = 128)
// ---------------------------------------------------------------------------
__global__ void kAggregate(const float* __restrict__ xsrc,
                           const int* __restrict__ src,
                           const int* __restrict__ dst,
                           int nE,
                           float* __restrict__ agg,
                           float* __restrict__ cnt) {
    int wid  = (int)(((size_t)blockIdx.x * blockDim.x + threadIdx.x) >> 5);
    int lane = threadIdx.x & 31;
    if (wid >= nE) return;
    int s = src[wid];
    int d = dst[wid];
    const float4 v =
        *reinterpret_cast<const float4*>(xsrc + (size_t)s * FDIM + lane * 4);
    float* ap = agg + (size_t)d * FDIM + lane * 4;
    atomicAdd(ap + 0, v.x);
    atomicAdd(ap + 1, v.y);
    atomicAdd(ap + 2, v.z);
    atomicAdd(ap + 3, v.w);
    if (lane == 0) atomicAdd(cnt + d, 1.0f);
}

// ---------------------------------------------------------------------------
// SAGE combine: out = relu?( mean @ Wl + bl + x_dst @ Wr )
// block = 8 waves; block owns a 16-row tile, wave owns a 16-col slab.
// Weights staged transposed in LDS once per block; all WMMA operands from LDS.
// Full-f32 WMMA (V_WMMA_F32_16X16X4_F32), 32 K-steps of 4, two wmma per step.
// ---------------------------------------------------------------------------
__global__ void __launch_bounds__(256) kSageGemm(
        const float* __restrict__ agg,
        const float* __restrict__ cnt,
        const float* __restrict__ xdst,
        const float* __restrict__ Wl,
        const float* __restrict__ bl,
        const float* __restrict__ Wr,
        float* __restrict__ out,
        int n_dst, int relu) {
    __shared__ float sWl[FDIM][WSTR];   // transposed: [col][k]
    __shared__ float sWr[FDIM][WSTR];
    __shared__ float sMean[16][LDSP];
    __shared__ float sX[16][LDSP];

    const int tid  = threadIdx.x;
    const int wave = tid >> 5;
    const int lane = tid & 31;
    const int m    = lane & 15;           // row within A-frag / N within B-frag
    const int koff = (lane >> 4) << 1;    // K offset: 0 (lanes 0-15) or 2
    const int col  = (wave << 4) | m;     // output column 0..127
    const int rowHi = (lane >> 4) << 3;   // C/D: M = i + 8*(lane>=16)

    // One-time transposed weight stage (global row-major [k][n] -> LDS [n][k]).
    for (int i = tid; i < FDIM * FDIM; i += 256) {
        int k = i >> 7, n = i & (FDIM - 1);
        sWl[n][k] = Wl[i];
        sWr[n][k] = Wr[i];
    }

    const float bias = bl[col];

    for (int mBase = blockIdx.x * 16; mBase < n_dst; mBase += gridDim.x * 16) {
        __syncthreads();
        // stage mean (agg/cnt) tile and x_dst tile into LDS
        for (int i = tid; i < 16 * FDIM; i += 256) {
            int r = i >> 7, c = i & (FDIM - 1);
            int gr = mBase + r;
            float cv = cnt[gr];
            float inv = 1.0f / fmaxf(cv, 1.0f);
            sMean[r][c] = agg[(size_t)gr * FDIM + c] * inv;
            sX[r][c]    = xdst[(size_t)gr * FDIM + c];
        }
        // hint-prefetch next tile's rows (global_prefetch_b8)
        int nextBase = mBase + gridDim.x * 16;
        if (nextBase < n_dst && tid < 16) {
            __builtin_prefetch(agg  + (size_t)(nextBase + tid) * FDIM, 0, 1);
            __builtin_prefetch(xdst + (size_t)(nextBase + tid) * FDIM, 0, 1);
        }
        __syncthreads();

        v8f acc;
#pragma unroll
        for (int i = 0; i < 8; ++i) acc[i] = bias;

#pragma unroll 4
        for (int kk = 0; kk < 32; ++kk) {
            const int k = (kk << 2) + koff;
            // mean @ Wl
            v2f a0;
            a0.x = sMean[m][k];
            a0.y = sMean[m][k + 1];
            v2f b0;
            b0.x = sWl[col][k];
            b0.y = sWl[col][k + 1];
            acc = __builtin_amdgcn_wmma_f32_16x16x4_f32(
                false, a0, false, b0, (short)0, acc, false, false);
            // x_dst @ Wr (same accumulator)
            v2f a1;
            a1.x = sX[m][k];
            a1.y = sX[m][k + 1];
            v2f b1;
            b1.x = sWr[col][k];
            b1.y = sWr[col][k + 1];
            acc = __builtin_amdgcn_wmma_f32_16x16x4_f32(
                false, a1, false, b1, (short)0, acc, false, false);
        }

        if (relu) {
#pragma unroll
            for (int i = 0; i < 8; ++i) acc[i] = fmaxf(acc[i], 0.0f);
        }
#pragma unroll
        for (int i = 0; i < 8; ++i) {
            int row = mBase + rowHi + i;
            out[(size_t)row * FDIM + col] = acc[i];
        }
    }
}

// ---------------------------------------------------------------------------
// edge decoder: out[q] = dot(z_srna[row[q]], z_mrna[col[q]]) -- wave per query
// ---------------------------------------------------------------------------
__global__ void kDecode(const float* __restrict__ zs,
                        const float* __restrict__ zm,
                        const int* __restrict__ rowq,
                        const int* __restrict__ colq,
                        int nQ, float* __restrict__ out) {
    int wid  = (int)(((size_t)blockIdx.x * blockDim.x + threadIdx.x) >> 5);
    int lane = threadIdx.x & 31;
    if (wid >= nQ) return;
    int r = rowq[wid];
    int c = colq[wid];
    const float4 a =
        *reinterpret_cast<const float4*>(zs + (size_t)r * FDIM + lane * 4);
    const float4 b =
        *reinterpret_cast<const float4*>(zm + (size_t)c * FDIM + lane * 4);
    float p = a.x * b.x + a.y * b.y + a.z * b.z + a.w * b.w;
#pragma unroll
    for (int off = 16; off > 0; off >>= 1) p += __shfl_down(p, off, 32);
    if (lane == 0) out[wid] = p;
}

// ---------------------------------------------------------------------------
static inline int imin_i(int a, int b) { return a < b ? a : b; }

extern "C" void kernel_launch(void* const* d_in, const int* in_sizes, int n_in,
                              void* d_out, int out_size, void* d_ws,
                              size_t ws_size, hipStream_t stream) {
    const float* x_srna = (const float*)d_in[0];
    const float* x_mrna = (const float*)d_in[1];
    const int* src_sm = (const int*)d_in[2];
    const int* dst_sm = (const int*)d_in[3];
    const int* src_ms = (const int*)d_in[4];
    const int* dst_ms = (const int*)d_in[5];
    const int* lbl_row = (const int*)d_in[6];
    const int* lbl_col = (const int*)d_in[7];
    const float* W1l_sm = (const float*)d_in[8];
    const float* W1r_sm = (const float*)d_in[9];
    const float* W1l_ms = (const float*)d_in[10];
    const float* W1r_ms = (const float*)d_in[11];
    const float* W2l_sm = (const float*)d_in[12];
    const float* W2r_sm = (const float*)d_in[13];
    const float* W2l_ms = (const float*)d_in[14];
    const float* W2r_ms = (const float*)d_in[15];
    const float* b1_sm = (const float*)d_in[16];
    const float* b1_ms = (const float*)d_in[17];
    const float* b2_sm = (const float*)d_in[18];
    const float* b2_ms = (const float*)d_in[19];

    const int nS = in_sizes[0] / FDIM;   // 20000
    const int nM = in_sizes[1] / FDIM;   // 100000
    const int E  = in_sizes[2];          // 1000000
    const int Q  = in_sizes[6];          // 500000

    float* ws = (float*)d_ws;
    float* h_srna = ws;
    float* h_mrna = h_srna + (size_t)nS * FDIM;
    float* z_srna = h_mrna + (size_t)nM * FDIM;
    float* z_mrna = z_srna + (size_t)nS * FDIM;
    float* agg    = z_mrna + (size_t)nM * FDIM;
    float* cnt    = agg    + (size_t)nM * FDIM;

    const int aggBlocks = (E + 7) / 8;           // 8 waves/block, 1 edge/wave
    const int decBlocks = (Q + 7) / 8;

    auto run_side = [&](const float* xsrc, const float* xdst,
                        const int* srcI, const int* dstI, int nDst,
                        const float* Wl, const float* bias, const float* Wr,
                        float* outp, int relu) {
        kZero<<<4096, 256, 0, stream>>>(agg, (size_t)nDst * FDIM);
        kZero<<<256, 256, 0, stream>>>(cnt, (size_t)nDst);
        kAggregate<<<aggBlocks, 256, 0, stream>>>(xsrc, srcI, dstI, E, agg, cnt);
        int tiles = nDst / 16;
        kSageGemm<<<imin_i(tiles, 1024), 256, 0, stream>>>(
            agg, cnt, xdst, Wl, bias, Wr, outp, nDst, relu);
    };

    // Layer 1 (+ReLU)
    run_side(x_srna, x_mrna, src_sm, dst_sm, nM, W1l_sm, b1_sm, W1r_sm, h_mrna, 1);
    run_side(x_mrna, x_srna, src_ms, dst_ms, nS, W1l_ms, b1_ms, W1r_ms, h_srna, 1);
    // Layer 2 (no activation)
    run_side(h_srna, h_mrna, src_sm, dst_sm, nM, W2l_sm, b2_sm, W2r_sm, z_mrna, 0);
    run_side(h_mrna, h_srna, src_ms, dst_ms, nS, W2l_ms, b2_ms, W2r_ms, z_srna, 0);
    // Edge decoder
    kDecode<<<decBlocks, 256, 0, stream>>>(z_srna, z_mrna, lbl_row, lbl_col, Q,
                                           (float*)d_out);
}